// PPI_DTI_Encoder_15367392985223
// MI455X (gfx1250) — compile-verified
//
#include <hip/hip_runtime.h>

// ---------------------------------------------------------------------------
// PPI->DTI GNN encoder for MI455X (gfx1250, wave32).
// GEMMs use native-f32 WMMA: v_wmma_f32_16x16x4_f32 (matches reference f32
// numerics; workload is scatter/atomic bound so low-precision WMMA gains
// nothing). Weight matrices are pre-packed pair-interleaved so each WMMA B
// fragment is one global_load_b64. Out-of-range rows are handled by CLAMPING
// the A-load row (output rows >= M are never stored, and WMMA row m depends
// only on A row m) -- keeps EXEC all-ones and the hot loop branch-free.
// Edge aggregation = float4 gathers + global_atomic_add_f32; the working set
// (~100MB) fits in the 192MB L2, so scatter atomics stay on-chip.
// ---------------------------------------------------------------------------

typedef __attribute__((ext_vector_type(2))) float v2f;
typedef __attribute__((ext_vector_type(8))) float v8f;

static constexpr int IN_F = 128;   // x_prot / x_drug feature width
static constexpr int D1   = 64;
static constexpr int D2   = 32;
static constexpr int OUTF = 64;

// ---------------- small elementwise helpers ----------------

__global__ void k_fill(float* __restrict__ p, float v, int n) {
  int i = blockIdx.x * blockDim.x + threadIdx.x;
  if (i < n) p[i] = v;
}

__global__ void k_deg_count(const int* __restrict__ dst, float* __restrict__ deg, int e) {
  int i = blockIdx.x * blockDim.x + threadIdx.x;
  if (i < e) atomicAdd(&deg[dst[i]], 1.0f);
}

__global__ void k_rsqrt_inplace(float* __restrict__ d, int n) {
  int i = blockIdx.x * blockDim.x + threadIdx.x;
  if (i < n) d[i] = rsqrtf(d[i]);
}

// Pack row-major B[K x N] into pair-interleaved P: P[(k>>1)*2N + n*2 + (k&1)].
// Makes each WMMA B fragment (two K-adjacent elements of one column) a single
// contiguous 8-byte load.
__global__ void k_pack_b(const float* __restrict__ B, float* __restrict__ P, int K, int N) {
  int i = blockIdx.x * blockDim.x + threadIdx.x;
  if (i >= K * N) return;
  int k = i / N, n = i - k * N;
  P[(k >> 1) * (2 * N) + n * 2 + (k & 1)] = B[i];
}

// ---------------- WMMA f32 GEMM: C[M x (NT*16)] = A[M x K] @ B[K x (NT*16)] --
// One wave32 per 16-row strip; NT accumulators reuse each A fragment.
// A fragment (16x4 f32): lane l -> row (l&15), k-pair kh = l>>4.
// B fragment (4x16 f32): lane l -> col (l&15), k-pair kh = l>>4.
// C/D (16x16 f32): vgpr v, lane l -> row v + 8*(l>>4), col (l&15).
// Bp is the pair-interleaved packed weight (see k_pack_b).

template <int NT, int K, bool GUARD>
__global__ void k_gemm_wmma(const float* __restrict__ A, const float* __restrict__ Bp,
                            float* __restrict__ C, int M) {
  constexpr int N = NT * 16;
  const int lane  = threadIdx.x;
  const int mbase = blockIdx.x * 16;
  const int mrow  = mbase + (lane & 15);
  const int kh    = lane >> 4;        // 0 or 1 -> k-pair select
  const int ncol  = lane & 15;
  // clamp (not predicate) OOB rows: branch-free, EXEC stays all-ones
  const int mload = GUARD ? min(mrow, M - 1) : mrow;

  v8f acc[NT] = {};
#pragma unroll
  for (int k = 0; k < K; k += 4) {
    v2f a = *(const v2f*)(A + (long long)mload * K + k + kh * 2);
    const float* bp = Bp + ((k >> 1) + kh) * (2 * N) + ncol * 2;
#pragma unroll
    for (int t = 0; t < NT; ++t) {
      v2f b = *(const v2f*)(bp + t * 32);
      acc[t] = __builtin_amdgcn_wmma_f32_16x16x4_f32(
          /*neg_a=*/false, a, /*neg_b=*/false, b,
          /*c_mod=*/(short)0, acc[t], /*reuse_a=*/false, /*reuse_b=*/false);
    }
  }

  const int mo = kh * 8;
#pragma unroll
  for (int t = 0; t < NT; ++t) {
#pragma unroll
    for (int v = 0; v < 8; ++v) {
      int m = mbase + mo + v;
      if (!GUARD || m < M) C[(long long)m * N + t * 16 + ncol] = acc[t][v];
    }
  }
}

// ---------------- GCN edge scatter: out[dst] += dinv[s]*dinv[d] * h[src] ----
// One thread per (edge, 4-feature group).

template <int F>
__global__ void k_gcn_scatter(const int* __restrict__ src, const int* __restrict__ dst,
                              const float* __restrict__ dinv, const float* __restrict__ h,
                              float* __restrict__ out, int e) {
  constexpr int G = F / 4;
  long long i = (long long)blockIdx.x * blockDim.x + threadIdx.x;
  long long total = (long long)e * G;
  if (i >= total) return;
  int ed = (int)(i / G);
  int g  = ((int)(i % G)) * 4;
  int s = src[ed], d = dst[ed];
  float nrm = dinv[s] * dinv[d];
  const float4 hv = *(const float4*)(h + (long long)s * F + g);
  float* o = out + (long long)d * F + g;
  atomicAdd(o + 0, nrm * hv.x);
  atomicAdd(o + 1, nrm * hv.y);
  atomicAdd(o + 2, nrm * hv.z);
  atomicAdd(o + 3, nrm * hv.w);
}

// self-loop term + bias (+ optional relu), in place on the aggregated buffer
template <int F, bool RELU>
__global__ void k_finalize(float* __restrict__ out, const float* __restrict__ htmp,
                           const float* __restrict__ dinv, const float* __restrict__ bias,
                           int n) {
  int i = blockIdx.x * blockDim.x + threadIdx.x;
  if (i >= n) return;
  int node = i / F;
  int f = i % F;
  float di = dinv[node];
  float v = out[i] + di * di * htmp[i] + bias[f];
  out[i] = RELU ? fmaxf(v, 0.0f) : v;
}

// ---------------- bipartite SAGE mean aggregation ----------------

__global__ void k_sage_scatter(const int* __restrict__ src, const int* __restrict__ dst,
                               const float* __restrict__ h, float* __restrict__ agg, int e) {
  constexpr int G = D2 / 4;
  long long i = (long long)blockIdx.x * blockDim.x + threadIdx.x;
  long long total = (long long)e * G;
  if (i >= total) return;
  int ed = (int)(i / G);
  int g  = ((int)(i % G)) * 4;
  int s = src[ed], d = dst[ed];
  const float4 hv = *(const float4*)(h + (long long)s * D2 + g);
  float* o = agg + (long long)d * D2 + g;
  atomicAdd(o + 0, hv.x);
  atomicAdd(o + 1, hv.y);
  atomicAdd(o + 2, hv.z);
  atomicAdd(o + 3, hv.w);
}

__global__ void k_cnt(const int* __restrict__ dst, float* __restrict__ cnt, int e) {
  int i = blockIdx.x * blockDim.x + threadIdx.x;
  if (i < e) atomicAdd(&cnt[dst[i]], 1.0f);
}

__global__ void k_mean(float* __restrict__ agg, const float* __restrict__ cnt, int n) {
  int i = blockIdx.x * blockDim.x + threadIdx.x;
  if (i >= n) return;
  agg[i] /= fmaxf(cnt[i / D2], 1.0f);
}

// ---------------- fused drug head: out = mean@Wl + bl + x_drug@Wr ----------
// Two K-loops accumulating into the same WMMA accumulators, packed weights.
// M=5000 is not a multiple of 16 -> clamp A-load rows, guard only stores.

__global__ void k_drug_gemm(const float* __restrict__ mean, const float* __restrict__ Wlp,
                            const float* __restrict__ bl, const float* __restrict__ xd,
                            const float* __restrict__ Wrp, float* __restrict__ out, int M) {
  constexpr int NT = OUTF / 16;  // 4
  const int lane  = threadIdx.x;
  const int mbase = blockIdx.x * 16;
  const int mrow  = mbase + (lane & 15);
  const int kh    = lane >> 4;
  const int ncol  = lane & 15;
  const int mload = min(mrow, M - 1);   // branch-free OOB handling

  v8f acc[NT] = {};

  // K = 32 over mean @ Wl
#pragma unroll
  for (int k = 0; k < D2; k += 4) {
    v2f a = *(const v2f*)(mean + (long long)mload * D2 + k + kh * 2);
    const float* bp = Wlp + ((k >> 1) + kh) * (2 * OUTF) + ncol * 2;
#pragma unroll
    for (int t = 0; t < NT; ++t) {
      v2f b = *(const v2f*)(bp + t * 32);
      acc[t] = __builtin_amdgcn_wmma_f32_16x16x4_f32(false, a, false, b, (short)0,
                                                     acc[t], false, false);
    }
  }
  // K = 128 over x_drug @ Wr
#pragma unroll
  for (int k = 0; k < IN_F; k += 4) {
    v2f a = *(const v2f*)(xd + (long long)mload * IN_F + k + kh * 2);
    const float* bp = Wrp + ((k >> 1) + kh) * (2 * OUTF) + ncol * 2;
#pragma unroll
    for (int t = 0; t < NT; ++t) {
      v2f b = *(const v2f*)(bp + t * 32);
      acc[t] = __builtin_amdgcn_wmma_f32_16x16x4_f32(false, a, false, b, (short)0,
                                                     acc[t], false, false);
    }
  }

  const int mo = kh * 8;
#pragma unroll
  for (int t = 0; t < NT; ++t) {
#pragma unroll
    for (int v = 0; v < 8; ++v) {
      int m = mbase + mo + v;
      if (m < M) out[(long long)m * OUTF + t * 16 + ncol] = acc[t][v] + bl[t * 16 + ncol];
    }
  }
}

// ---------------------------------------------------------------------------

static inline int cdiv(long long a, long long b) { return (int)((a + b - 1) / b); }

extern "C" void kernel_launch(void* const* d_in, const int* in_sizes, int n_in,
                              void* d_out, int out_size, void* d_ws, size_t ws_size,
                              hipStream_t stream) {
  const float* x_prot = (const float*)d_in[0];
  const float* x_drug = (const float*)d_in[1];
  const int*   pp     = (const int*)d_in[2];
  const int*   pd_src = (const int*)d_in[3];
  const int*   pd_dst = (const int*)d_in[4];
  const float* W1 = (const float*)d_in[5];
  const float* b1 = (const float*)d_in[6];
  const float* W2 = (const float*)d_in[7];
  const float* b2 = (const float*)d_in[8];
  const float* Wl = (const float*)d_in[9];
  const float* bl = (const float*)d_in[10];
  const float* Wr = (const float*)d_in[11];

  const int n_prot = in_sizes[0] / IN_F;
  const int n_drug = in_sizes[1] / IN_F;
  const int e_pp   = in_sizes[2] / 2;
  const int e_pd   = in_sizes[3];
  const int* pp_src = pp;
  const int* pp_dst = pp + e_pp;

  // workspace layout (floats)
  float* ws = (float*)d_ws;
  size_t off = 0;
  float* dinv  = ws + off; off += (size_t)n_prot;
  float* h1tmp = ws + off; off += (size_t)n_prot * D1;
  float* h1    = ws + off; off += (size_t)n_prot * D1;
  float* h2tmp = ws + off; off += (size_t)n_prot * D2;
  float* agg3  = ws + off; off += (size_t)n_drug * D2;
  float* cnt   = ws + off; off += (size_t)n_drug;
  float* W1p   = ws + off; off += (size_t)IN_F * D1;   // packed weights
  float* W2p   = ws + off; off += (size_t)D1 * D2;
  float* Wlp   = ws + off; off += (size_t)D2 * OUTF;
  float* Wrp   = ws + off; off += (size_t)IN_F * OUTF;

  float* h2   = (float*)d_out;                          // output 0: [n_prot, 32]
  float* dout = (float*)d_out + (size_t)n_prot * D2;    // output 1: [n_drug, 64]

  const int T = 256;

  // pack weight matrices (pair-interleaved for single-b64 WMMA B fragments)
  k_pack_b<<<cdiv(IN_F * D1, T), T, 0, stream>>>(W1, W1p, IN_F, D1);
  k_pack_b<<<cdiv(D1 * D2, T), T, 0, stream>>>(W2, W2p, D1, D2);
  k_pack_b<<<cdiv(D2 * OUTF, T), T, 0, stream>>>(Wl, Wlp, D2, OUTF);
  k_pack_b<<<cdiv(IN_F * OUTF, T), T, 0, stream>>>(Wr, Wrp, IN_F, OUTF);

  // degrees (init 1.0 covers the self-loop), then dinv = rsqrt(deg)
  k_fill<<<cdiv(n_prot, T), T, 0, stream>>>(dinv, 1.0f, n_prot);
  k_deg_count<<<cdiv(e_pp, T), T, 0, stream>>>(pp_dst, dinv, e_pp);
  k_rsqrt_inplace<<<cdiv(n_prot, T), T, 0, stream>>>(dinv, n_prot);

  // --- GCN layer 1: h1 = relu(scatter(norm * (x@W1)) + dinv^2*(x@W1) + b1)
  if (n_prot % 16 == 0)
    k_gemm_wmma<D1 / 16, IN_F, false><<<cdiv(n_prot, 16), 32, 0, stream>>>(x_prot, W1p, h1tmp, n_prot);
  else
    k_gemm_wmma<D1 / 16, IN_F, true><<<cdiv(n_prot, 16), 32, 0, stream>>>(x_prot, W1p, h1tmp, n_prot);
  hipMemsetAsync(h1, 0, (size_t)n_prot * D1 * sizeof(float), stream);
  k_gcn_scatter<D1><<<cdiv((long long)e_pp * (D1 / 4), T), T, 0, stream>>>(
      pp_src, pp_dst, dinv, h1tmp, h1, e_pp);
  k_finalize<D1, true><<<cdiv((long long)n_prot * D1, T), T, 0, stream>>>(
      h1, h1tmp, dinv, b1, n_prot * D1);

  // --- GCN layer 2: h2 (written straight into d_out segment 0)
  if (n_prot % 16 == 0)
    k_gemm_wmma<D2 / 16, D1, false><<<cdiv(n_prot, 16), 32, 0, stream>>>(h1, W2p, h2tmp, n_prot);
  else
    k_gemm_wmma<D2 / 16, D1, true><<<cdiv(n_prot, 16), 32, 0, stream>>>(h1, W2p, h2tmp, n_prot);
  hipMemsetAsync(h2, 0, (size_t)n_prot * D2 * sizeof(float), stream);
  k_gcn_scatter<D2><<<cdiv((long long)e_pp * (D2 / 4), T), T, 0, stream>>>(
      pp_src, pp_dst, dinv, h2tmp, h2, e_pp);
  k_finalize<D2, false><<<cdiv((long long)n_prot * D2, T), T, 0, stream>>>(
      h2, h2tmp, dinv, b2, n_prot * D2);

  // --- bipartite SAGE mean aggregation
  hipMemsetAsync(agg3, 0, ((size_t)n_drug * D2 + n_drug) * sizeof(float), stream);
  k_sage_scatter<<<cdiv((long long)e_pd * (D2 / 4), T), T, 0, stream>>>(
      pd_src, pd_dst, h2, agg3, e_pd);
  k_cnt<<<cdiv(e_pd, T), T, 0, stream>>>(pd_dst, cnt, e_pd);
  k_mean<<<cdiv((long long)n_drug * D2, T), T, 0, stream>>>(agg3, cnt, n_drug * D2);

  // --- fused drug head: dout = mean@Wl + bl + x_drug@Wr
  k_drug_gemm<<<cdiv(n_drug, 16), 32, 0, stream>>>(agg3, Wlp, bl, x_drug, Wrp, dout, n_drug);
}